// LogTrans_70454643523785
// MI455X (gfx1250) — compile-verified
//
#include <hip/hip_runtime.h>

typedef float v2f __attribute__((ext_vector_type(2)));
typedef float v8f __attribute__((ext_vector_type(8)));

#define SQ   4096      // sequence length
#define DH   64        // head dim
#define NK   12        // gathered keys = log2(SQ)
#define NBH  32        // B*H

// One wave (32 lanes) computes a 16-query tile of log-sparse attention for one (b,h).
// Block = 256 threads = 8 waves = 128 queries; grid = (B*H, SQ/128).
__global__ __launch_bounds__(256) void logattn_wmma_kernel(
    const float* __restrict__ Q, const float* __restrict__ K,
    const float* __restrict__ V, const unsigned char* __restrict__ mask,
    float* __restrict__ out_ctx, float* __restrict__ out_attn)
{
    __shared__ float lds_attn[8][16 * 16];   // per-wave 16x16 attn transpose tile

    const int bh    = blockIdx.x;            // (b*H + h)
    const int wave  = threadIdx.x >> 5;
    const int lane  = threadIdx.x & 31;
    const int half  = lane >> 4;             // 0: K={0,1}, 1: K={2,3} (WMMA f32 16x16x4 layout)
    const int l16   = lane & 15;
    const int dpair = half * 2;
    const int q0    = blockIdx.y * 128 + wave * 16;

    // ---- K_g into score-B register layout (B[k][n] = Kg[n][k], lane l16 = key n) ----
    const bool kvalid = (l16 < NK);
    const int  indn   = kvalid ? (SQ - (SQ >> l16)) : 0;   // ind(k) = SQ - 2^(12-k)
    const float* Krow = K + ((size_t)bh * SQ + indn) * DH;

    v2f bs[16];
    #pragma unroll
    for (int c = 0; c < 16; ++c) {
        v2f t = *(const v2f*)(Krow + 4 * c + dpair);
        bs[c].x = kvalid ? t.x : 0.0f;
        bs[c].y = kvalid ? t.y : 0.0f;
    }

    // ---- V_g into context-B register layout: tile t (d0=16t), chunk kc: B[k][n]=Vg[k][16t+n] ----
    v2f bv[4][3];
    #pragma unroll
    for (int t = 0; t < 4; ++t) {
        #pragma unroll
        for (int kc = 0; kc < 3; ++kc) {
            const int k0 = 4 * kc + dpair;
            const int i0 = SQ - (SQ >> k0);
            const int i1 = SQ - (SQ >> (k0 + 1));
            bv[t][kc].x = V[((size_t)bh * SQ + i0) * DH + 16 * t + l16];
            bv[t][kc].y = V[((size_t)bh * SQ + i1) * DH + 16 * t + l16];
        }
    }

    // ---- scores: 16 chained V_WMMA_F32_16X16X4_F32 over D=64 ----
    const float* Qrow = Q + ((size_t)bh * SQ + q0 + l16) * DH;
    v8f c = {};
    #pragma unroll
    for (int cc = 0; cc < 16; ++cc) {
        v2f a = *(const v2f*)(Qrow + 4 * cc + dpair);
        c = __builtin_amdgcn_wmma_f32_16x16x4_f32(false, a, false, bs[cc],
                                                  (short)0, c, false, false);
    }

    // ---- scale + mask (score layout: VGPR r -> query q0+r+half*8, lane l16 -> key) ----
    float sc[8];
    const size_t mbase = (size_t)bh * SQ * SQ;
    #pragma unroll
    for (int r = 0; r < 8; ++r) {
        const int q = q0 + r + half * 8;
        float s = c[r] * 0.125f;                               // 1/sqrt(64)
        const unsigned char m = mask[mbase + (size_t)q * SQ + indn];
        s = (m != 0) ? -1.0e9f : s;                            // where(mask, -1e9, s)
        sc[r] = kvalid ? s : -3.0e38f;                         // pad keys vanish
    }

    // ---- softmax over 16-lane key groups ----
    #pragma unroll
    for (int r = 0; r < 8; ++r) {
        float m = sc[r];
        #pragma unroll
        for (int off = 8; off >= 1; off >>= 1) m = fmaxf(m, __shfl_xor(m, off, 16));
        const float e = __expf(sc[r] - m);
        float s = e;
        #pragma unroll
        for (int off = 8; off >= 1; off >>= 1) s += __shfl_xor(s, off, 16);
        sc[r] = e / s;
    }

    // ---- write attn output + stage transpose tile in LDS ----
    float* lw = lds_attn[wave];
    float* oa = out_attn + (size_t)bh * SQ * NK;
    #pragma unroll
    for (int r = 0; r < 8; ++r) {
        const int ql = r + half * 8;
        lw[ql * 16 + l16] = sc[r];
        if (kvalid) oa[(size_t)(q0 + ql) * NK + l16] = sc[r];
    }
    __syncthreads();   // uniform across block; orders LDS store -> load

    // ---- attn in A-layout from LDS: a[kc] = attn[q=l16][4kc+dpair .. +1] ----
    v2f aa[3];
    #pragma unroll
    for (int kc = 0; kc < 3; ++kc)
        aa[kc] = *(const v2f*)(lw + l16 * 16 + 4 * kc + dpair);

    // ---- context: 4 d-tiles x 3 k-chunks of WMMA, then store ----
    float* oc = out_ctx + (size_t)bh * SQ * DH;
    #pragma unroll
    for (int t = 0; t < 4; ++t) {
        v8f d = {};
        #pragma unroll
        for (int kc = 0; kc < 3; ++kc)
            d = __builtin_amdgcn_wmma_f32_16x16x4_f32(false, aa[kc], false, bv[t][kc],
                                                      (short)0, d, false, false);
        #pragma unroll
        for (int r = 0; r < 8; ++r) {
            const int q = q0 + r + half * 8;
            oc[(size_t)q * DH + 16 * t + l16] = d[r];
        }
    }
}

extern "C" void kernel_launch(void* const* d_in, const int* in_sizes, int n_in,
                              void* d_out, int out_size, void* d_ws, size_t ws_size,
                              hipStream_t stream) {
    const float* Q = (const float*)d_in[0];
    const float* K = (const float*)d_in[1];
    const float* V = (const float*)d_in[2];
    const unsigned char* mask = (const unsigned char*)d_in[3];   // jnp.bool_ -> 1 byte

    float* out_ctx  = (float*)d_out;                              // (4,8,4096,64)
    float* out_attn = out_ctx + (size_t)NBH * SQ * DH;            // (4,8,4096,12)

    dim3 grid(NBH, SQ / 128);   // 32 x 32 blocks
    logattn_wmma_kernel<<<grid, 256, 0, stream>>>(Q, K, V, mask, out_ctx, out_attn);
}